// RelGATLayerSemantic_63230508531931
// MI455X (gfx1250) — compile-verified
//
#include <hip/hip_runtime.h>

typedef __attribute__((ext_vector_type(2))) float v2f;
typedef __attribute__((ext_vector_type(8))) float v8f;

#define HEADS 4
#define DK 32
#define IN_DIM 256
#define OUT_DIM 128
#define NEG_SLOPE 0.2f
#define EPS 1e-12f
#define LDS_STRIDE 260  // 256 + 4 pad: A-frag reads walk rows -> spread across banks

// ---------- helpers: monotone float<->uint key for atomicMax on floats ----------
__device__ __forceinline__ unsigned fkey(float f) {
    unsigned u = __float_as_uint(f);
    return (u & 0x80000000u) ? ~u : (u | 0x80000000u);
}
__device__ __forceinline__ float funkey(unsigned k) {
    return (k & 0x80000000u) ? __uint_as_float(k ^ 0x80000000u) : __uint_as_float(~k);
}

// ---------- 0) zero accumulators (out_h, denom) and max keys ----------
__global__ void init_kernel(float* __restrict__ outH, float* __restrict__ denom,
                            unsigned* __restrict__ gmaxk, int nOut, int nDen) {
    int i = blockIdx.x * blockDim.x + threadIdx.x;
    int stride = gridDim.x * blockDim.x;
    for (int j = i; j < nOut; j += stride) outH[j] = 0.0f;
    for (int j = i; j < nDen; j += stride) denom[j] = 0.0f;
    if (i < HEADS) gmaxk[i] = 0u;  // key 0 < key(any float)
}

// ---------- 1) H_proj = H @ W.T via V_WMMA_F32_16X16X4_F32 ----------
// grid.x = N/16 (M tiles), block = 256 (8 waves), wave w -> col tile w.
// The 16x256 A-tile is staged into LDS ONCE per block with async LDS loads
// (ASYNCcnt path), instead of each of the 8 waves re-reading it from global.
__global__ void proj_wmma_kernel(const float* __restrict__ H, const float* __restrict__ W,
                                 float* __restrict__ P) {
    __shared__ float sH[16 * LDS_STRIDE];  // padded 16x256 tile (~16.6 KB)

    const int tid   = threadIdx.x;
    const int lane  = tid & 31;
    const int wave  = tid >> 5;              // 0..7 = output col tile
    const int mtile = blockIdx.x;
    const int l15   = lane & 15;
    const int khalf = (lane >> 4) << 1;      // 0 or 2

    // --- async stage: 16 rows x 256 cols f32 = 16384 B; 256 thr x 16 B x 4 passes ---
    const float* __restrict__ gtile = H + (size_t)(mtile * 16) * IN_DIM;
#pragma unroll
    for (int it = 0; it < 4; ++it) {
        const int f   = it * 1024 + tid * 4;        // flat float idx in 16x256 tile
        const int r   = f >> 8;                     // row 0..15
        const int cix = f & 255;                    // col
        unsigned loff = (unsigned)(uintptr_t)(&sH[r * LDS_STRIDE + cix]);
        const float* gp = gtile + f;
        asm volatile("global_load_async_to_lds_b128 %0, %1, off"
                     :: "v"(loff), "v"(gp) : "memory");
    }
    asm volatile("s_wait_asynccnt 0" ::: "memory");
    __syncthreads();

    const int col = wave * 16 + l15;                // B: N = lane%16
    const float* __restrict__ wcol = W + (size_t)col * IN_DIM;
    const float* __restrict__ arow = &sH[l15 * LDS_STRIDE];  // A: M = lane%16

    v8f c = {0.f, 0.f, 0.f, 0.f, 0.f, 0.f, 0.f, 0.f};
#pragma unroll 8
    for (int kb = 0; kb < IN_DIM; kb += 4) {
        v2f a, b;
        a.x = arow[kb + khalf];
        a.y = arow[kb + khalf + 1];
        b.x = wcol[kb + khalf];
        b.y = wcol[kb + khalf + 1];
        c = __builtin_amdgcn_wmma_f32_16x16x4_f32(false, a, false, b, (short)0, c,
                                                  false, false);
    }
    const int mbase = mtile * 16 + 8 * (lane >> 4);  // D: M = v + 8*(lane/16)
#pragma unroll
    for (int v = 0; v < 8; ++v) {
        P[(size_t)(mbase + v) * OUT_DIM + col] = c[v];
    }
}

// ---------- 2) per-edge logits + LeakyReLU + global per-head max ----------
__global__ void logits_kernel(const float* __restrict__ P, const long long* __restrict__ eidx,
                              const float* __restrict__ sim, const float* __restrict__ a_src,
                              const float* __restrict__ a_dst, float* __restrict__ ebuf,
                              unsigned* __restrict__ gmaxk, int E) {
    __shared__ float sA[OUT_DIM];
    __shared__ float sB[OUT_DIM];
    __shared__ float red[256];
    const int tid = threadIdx.x;
    if (tid < OUT_DIM) { sA[tid] = a_src[tid]; sB[tid] = a_dst[tid]; }
    __syncthreads();

    float lmax[HEADS] = {-3.4e38f, -3.4e38f, -3.4e38f, -3.4e38f};
    const int stride = gridDim.x * blockDim.x;
    for (int e = blockIdx.x * blockDim.x + tid; e < E; e += stride) {
        if (e + stride < E) {  // hint the next streamed elements into cache
            __builtin_prefetch(&eidx[e + stride], 0, 1);
            __builtin_prefetch(&eidx[E + e + stride], 0, 1);
            __builtin_prefetch(&sim[e + stride], 0, 1);
        }
        const long long s = eidx[e];
        const long long d = eidx[E + e];
        const float* __restrict__ ps = P + (size_t)s * OUT_DIM;
        const float* __restrict__ pd = P + (size_t)d * OUT_DIM;
        const float se = sim[e];  // BETA == 1.0
#pragma unroll
        for (int h = 0; h < HEADS; ++h) {
            float acc = se;
#pragma unroll
            for (int k = 0; k < DK; ++k) {
                acc += ps[h * DK + k] * sA[h * DK + k];
                acc += pd[h * DK + k] * sB[h * DK + k];
            }
            acc = (acc >= 0.0f) ? acc : NEG_SLOPE * acc;  // LeakyReLU(0.2)
            ebuf[(size_t)e * HEADS + h] = acc;
            lmax[h] = fmaxf(lmax[h], acc);
        }
    }
    // block reduction per head, then one atomicMax per head per block
    for (int h = 0; h < HEADS; ++h) {
        red[tid] = lmax[h];
        __syncthreads();
        for (int off = 128; off > 0; off >>= 1) {
            if (tid < off) red[tid] = fmaxf(red[tid], red[tid + off]);
            __syncthreads();
        }
        if (tid == 0) atomicMax(&gmaxk[h], fkey(red[0]));
        __syncthreads();
    }
}

// ---------- 3) exp(e - max) + segment-sum denominator over dst ----------
__global__ void norm_kernel(const long long* __restrict__ eidx, float* __restrict__ ebuf,
                            float* __restrict__ denom, const unsigned* __restrict__ gmaxk,
                            int E) {
    float gm[HEADS];
#pragma unroll
    for (int h = 0; h < HEADS; ++h) gm[h] = funkey(gmaxk[h]);
    const int stride = gridDim.x * blockDim.x;
    for (int e = blockIdx.x * blockDim.x + threadIdx.x; e < E; e += stride) {
        const long long d = eidx[E + e];
#pragma unroll
        for (int h = 0; h < HEADS; ++h) {
            float x = __expf(ebuf[(size_t)e * HEADS + h] - gm[h]);
            ebuf[(size_t)e * HEADS + h] = x;
            atomicAdd(&denom[(size_t)d * HEADS + h], x);
        }
    }
}

// ---------- 4) alpha, alpha_mean, scatter alpha*h_src into out_h (wave per edge) ----------
__global__ void scatter_kernel(const float* __restrict__ P, const long long* __restrict__ eidx,
                               const float* __restrict__ ebuf, const float* __restrict__ denom,
                               float* __restrict__ outH, float* __restrict__ alpha_mean, int E) {
    const int lane = threadIdx.x & 31;
    const int wid  = (blockIdx.x * blockDim.x + threadIdx.x) >> 5;
    const int nw   = (gridDim.x * blockDim.x) >> 5;
    for (int e = wid; e < E; e += nw) {
        const long long s = eidx[e];
        const long long d = eidx[E + e];
        float alpha[HEADS];
        float am = 0.0f;
#pragma unroll
        for (int h = 0; h < HEADS; ++h) {
            float a = ebuf[(size_t)e * HEADS + h] /
                      (denom[(size_t)d * HEADS + h] + EPS);
            alpha[h] = a;
            am += a;
        }
        if (lane == 0) alpha_mean[e] = am * 0.25f;
        const float* __restrict__ ps = P + (size_t)s * OUT_DIM;
        float* __restrict__ po = outH + (size_t)d * OUT_DIM;
#pragma unroll
        for (int h = 0; h < HEADS; ++h) {
            const int c = h * DK + lane;  // coalesced across lanes
            atomicAdd(&po[c], alpha[h] * ps[c]);
        }
    }
}

// ---------- 5) msg_base = mean_h(h_src) @ msg_lin_w.T via WMMA ----------
// wave per 16-edge group (grid-stride); B fragments (32x32 weights) live in regs.
__global__ void msg_wmma_kernel(const float* __restrict__ P, const long long* __restrict__ eidx,
                                const float* __restrict__ mlw, float* __restrict__ msg, int E) {
    const int lane  = threadIdx.x & 31;
    const int wid   = (blockIdx.x * blockDim.x + threadIdx.x) >> 5;
    const int nw    = (gridDim.x * blockDim.x) >> 5;
    const int l15   = lane & 15;
    const int khalf = (lane >> 4) << 1;
    const int nGroups = E >> 4;  // E % 16 == 0

    // B[k][o] = mlw[o*32 + k]; fragment: N = lane%16, K = 4*i + khalf + v
    v2f bf[8][2];
#pragma unroll
    for (int i = 0; i < 8; ++i) {
#pragma unroll
        for (int t = 0; t < 2; ++t) {
            const int o = 16 * t + l15;
            const int k = 4 * i + khalf;
            bf[i][t].x = mlw[o * DK + k];
            bf[i][t].y = mlw[o * DK + k + 1];
        }
    }

    for (int g = wid; g < nGroups; g += nw) {
        const int base = g * 16;
        const long long r = eidx[base + l15];  // src row for this lane's A rows
        const float* __restrict__ p = P + (size_t)r * OUT_DIM;
        v8f c0 = {0.f, 0.f, 0.f, 0.f, 0.f, 0.f, 0.f, 0.f};
        v8f c1 = {0.f, 0.f, 0.f, 0.f, 0.f, 0.f, 0.f, 0.f};
#pragma unroll
        for (int i = 0; i < 8; ++i) {
            const int k = 4 * i + khalf;
            v2f a;
            a.x = 0.25f * (p[k]     + p[DK + k]     + p[2*DK + k]     + p[3*DK + k]);
            a.y = 0.25f * (p[k + 1] + p[DK + k + 1] + p[2*DK + k + 1] + p[3*DK + k + 1]);
            c0 = __builtin_amdgcn_wmma_f32_16x16x4_f32(false, a, false, bf[i][0],
                                                       (short)0, c0, false, false);
            c1 = __builtin_amdgcn_wmma_f32_16x16x4_f32(false, a, false, bf[i][1],
                                                       (short)0, c1, false, false);
        }
        const int mbase = base + 8 * (lane >> 4);
#pragma unroll
        for (int v = 0; v < 8; ++v) {
            const size_t eo = (size_t)(mbase + v) * DK;
            msg[eo + l15]      = c0[v];
            msg[eo + 16 + l15] = c1[v];
        }
    }
}

extern "C" void kernel_launch(void* const* d_in, const int* in_sizes, int n_in,
                              void* d_out, int out_size, void* d_ws, size_t ws_size,
                              hipStream_t stream) {
    const float*     H     = (const float*)d_in[0];
    const long long* eidx  = (const long long*)d_in[1];
    const float*     sim   = (const float*)d_in[2];
    const float*     W     = (const float*)d_in[3];
    const float*     a_src = (const float*)d_in[4];
    const float*     a_dst = (const float*)d_in[5];
    const float*     mlw   = (const float*)d_in[6];

    const int N = in_sizes[0] / IN_DIM;   // 50000
    const int E = in_sizes[2];            // 800000

    // workspace layout
    float*    P     = (float*)d_ws;                      // [N,128] projected features
    float*    ebuf  = P + (size_t)N * OUT_DIM;           // [E,4] logits -> exp
    float*    denom = ebuf + (size_t)E * HEADS;          // [N,4]
    unsigned* gmaxk = (unsigned*)(denom + (size_t)N * HEADS);  // [4]

    // output layout (concat in return order)
    float* outH       = (float*)d_out;                   // [N,128]
    float* alpha_mean = outH + (size_t)N * OUT_DIM;      // [E]
    float* msg        = alpha_mean + (size_t)E;          // [E,32]

    // 0) zero accumulators
    {
        int total = N * OUT_DIM;
        int blocks = (total + 255) / 256;
        if (blocks > 4096) blocks = 4096;
        init_kernel<<<blocks, 256, 0, stream>>>(outH, denom, gmaxk, total, N * HEADS);
    }
    // 1) projection GEMM (WMMA f32, async-LDS staged A tile)
    proj_wmma_kernel<<<N / 16, 256, 0, stream>>>(H, W, P);
    // 2) logits + global max
    logits_kernel<<<2048, 256, 0, stream>>>(P, eidx, sim, a_src, a_dst, ebuf, gmaxk, E);
    // 3) exp + denom segment-sum
    norm_kernel<<<2048, 256, 0, stream>>>(eidx, ebuf, denom, gmaxk, E);
    // 4) alpha + scatter-add out_h (wave per edge)
    scatter_kernel<<<4096, 256, 0, stream>>>(P, eidx, ebuf, denom, outH, alpha_mean, E);
    // 5) msg_base GEMM (WMMA f32, wave per 16-edge tile)
    msg_wmma_kernel<<<2048, 256, 0, stream>>>(P, eidx, mlw, msg, E);
}